// Generator_9036611191160
// MI455X (gfx1250) — compile-verified
//
#include <hip/hip_runtime.h>
#include <hip/hip_bf16.h>
#include <math.h>

// Problem constants from the reference
#define BATCH 64
#define CH    3
#define IH    512
#define IW    512
#define PLANE (IH * IW)

// 8-byte pair with only 4-byte alignment guarantee (xb is an arbitrary int).
// Backend emits a single global_load_b64 (unaligned-access mode).
typedef struct __attribute__((packed, aligned(4))) { float lo, hi; } fpair;

// ---------------------------------------------------------------------------
// Main kernel (placed FIRST so the disasm snippet shows it): one thread = one
// output pixel (all 3 channels).
//  - params indexed by blockIdx.z only -> uniform address -> s_load_b128
//  - paired-corner b64 gathers at clamped base, zero-padding folded into weights
//  - nontemporal stores: streaming 201MB output must not evict L2-resident
//    input image planes (3MB/batch, fully L2-resident)
// ---------------------------------------------------------------------------
__global__ __launch_bounds__(256) void affine_sample_kernel(
    const float* __restrict__ img,
    const float4* __restrict__ params,
    float* __restrict__ out) {
  const int b   = blockIdx.z;
  const int pix = blockIdx.x * 256 + threadIdx.x; // 0 .. H*W-1
  const int y   = pix >> 9;                       // /512
  const int x   = pix & (IW - 1);

  const float4 p = params[b]; // uniform -> scalar load

  // align_corners=False pixel-center normalized coords
  const float gx = (2.0f * (float)x + 1.0f) * (1.0f / (float)IW) - 1.0f;
  const float gy = (2.0f * (float)y + 1.0f) * (1.0f / (float)IH) - 1.0f;

  // theta * [gx, gy, 1]
  const float u = fmaf(p.x, gx, fmaf(-p.y, gy, p.z));
  const float v = fmaf(p.y, gx, fmaf( p.x, gy, p.w));

  // normalized -> input pixel space: ((u+1)*W - 1) * 0.5
  const float ixf = fmaf(u + 1.0f, 0.5f * (float)IW, -0.5f);
  const float iyf = fmaf(v + 1.0f, 0.5f * (float)IH, -0.5f);

  const float fx0 = floorf(ixf);
  const float fy0 = floorf(iyf);
  const int x0 = (int)fx0;
  const int y0 = (int)fy0;

  float wx1 = ixf - fx0;
  float wy1 = iyf - fy0;
  float wx0 = 1.0f - wx1;
  float wy0 = 1.0f - wy1;

  // zero-padding: fold validity into weights
  wx0 = (x0     >= 0 && x0     < IW) ? wx0 : 0.0f;
  wx1 = (x0 + 1 >= 0 && x0 + 1 < IW) ? wx1 : 0.0f;
  wy0 = (y0     >= 0 && y0     < IH) ? wy0 : 0.0f;
  wy1 = (y0 + 1 >= 0 && y0 + 1 < IH) ? wy1 : 0.0f;

  // clamped pair base so both elements of the b64 are always in-bounds
  const int xb  = min(max(x0, 0), IW - 2);
  const int x0c = min(max(x0,     0), IW - 1);
  const int x1c = min(max(x0 + 1, 0), IW - 1);
  const int y0c = min(max(y0,     0), IH - 1);
  const int y1c = min(max(y0 + 1, 0), IH - 1);

  // Map each logical corner's weight onto the low/high element of the pair.
  // (Weights of out-of-range corners are already zero, so clamping is safe.)
  const float wlo = ((x0c == xb) ? wx0 : 0.0f) + ((x1c == xb) ? wx1 : 0.0f);
  const float whi = ((x0c != xb) ? wx0 : 0.0f) + ((x1c != xb) ? wx1 : 0.0f);

  const float* base = img + (size_t)b * (CH * PLANE);
  const size_t o0 = (size_t)y0c * IW + (size_t)xb;
  const size_t o1 = (size_t)y1c * IW + (size_t)xb;

  float acc[CH];
#pragma unroll
  for (int c = 0; c < CH; ++c) {
    const float* plane = base + (size_t)c * PLANE;
    const fpair r0 = *(const fpair*)(plane + o0); // global_load_b64
    const fpair r1 = *(const fpair*)(plane + o1); // global_load_b64
    const float top = fmaf(r0.lo, wlo, r0.hi * whi);
    const float bot = fmaf(r1.lo, wlo, r1.hi * whi);
    acc[c] = fmaf(top, wy0, bot * wy1);
  }

  float* obase = out + (size_t)b * (CH * PLANE) + (size_t)pix;
#pragma unroll
  for (int c = 0; c < CH; ++c) {
    // streaming output: NT temporal hint keeps L2 for the input images
    __builtin_nontemporal_store(acc[c], obase + (size_t)c * PLANE);
  }
}

// ---------------------------------------------------------------------------
// Tiny kernel: per-batch affine params [ca, sa, tx, ty] = [cos/s, sin/s, tx/s, ty/s]
// Fast hw sin/cos: rot ~ N(0,1), well inside v_sin/v_cos valid range.
// ---------------------------------------------------------------------------
__global__ __launch_bounds__(64) void theta_kernel(
    const float* __restrict__ rot,
    const float* __restrict__ trans,
    const float* __restrict__ scale,
    float4* __restrict__ params) {
  const int b = threadIdx.x; // exactly 64 threads launched
  const float s  = scale[b] + 1e-18f;
  const float is = 1.0f / s;
  const float r  = rot[b];
  float4 p;
  p.x = __cosf(r) * is;          // ca
  p.y = __sinf(r) * is;          // sa
  p.z = trans[2 * b + 0] * is;   // tx
  p.w = trans[2 * b + 1] * is;   // ty
  params[b] = p;
}

// ---------------------------------------------------------------------------
// Host-side launcher
// d_in order: images [B,C,H,W] f32, rot [B] f32, trans [B,2] f32, scale [B] f32
// d_out: [B,C,H,W] f32
// ---------------------------------------------------------------------------
extern "C" void kernel_launch(void* const* d_in, const int* in_sizes, int n_in,
                              void* d_out, int out_size, void* d_ws, size_t ws_size,
                              hipStream_t stream) {
  const float* images = (const float*)d_in[0];
  const float* rot    = (const float*)d_in[1];
  const float* trans  = (const float*)d_in[2];
  const float* scale  = (const float*)d_in[3];
  float*       outp   = (float*)d_out;
  float4*      params = (float4*)d_ws; // 64 * 16B = 1 KB scratch

  theta_kernel<<<1, 64, 0, stream>>>(rot, trans, scale, params);

  dim3 grid(PLANE / 256, 1, BATCH); // 1024 x 1 x 64
  dim3 block(256, 1, 1);
  affine_sample_kernel<<<grid, block, 0, stream>>>(images, params, outp);
}